// SDE_74964359184471
// MI455X (gfx1250) — compile-verified
//
#include <hip/hip_runtime.h>
#include <hip/hip_bf16.h>

#define AS1 __attribute__((address_space(1)))
#define AS3 __attribute__((address_space(3)))

typedef int v4i __attribute__((ext_vector_type(4)));

namespace {
constexpr int BATCH  = 16384;
constexpr int TSIZE  = 4096;
constexpr int NSTEPS = TSIZE - 1;                       // 4095 EM steps
constexpr int COLS   = 128;                             // trajectories per WG (4 wave32 waves)
constexpr int NWG    = BATCH / COLS;                    // 128 workgroups -> more WGPs engaged
constexpr int CHUNK  = 16;                              // time steps per LDS buffer
constexpr int NBUF   = 6;                               // pipeline depth (prefetch 5 ahead)
constexpr int ROWS_PER_CHUNK   = 2 * CHUNK;             // 32 rows of COLS floats
constexpr int FLOATS_PER_CHUNK = ROWS_PER_CHUNK * COLS; // 4096 floats = 16 KB
constexpr int VECS_PER_CHUNK   = FLOATS_PER_CHUNK / 4;  // 1024 b128 transfers
constexpr int VECS_PER_THREAD  = VECS_PER_CHUNK / COLS; // 8 async instr / wave / chunk
constexpr int NCHUNK = (NSTEPS + CHUNK - 1) / CHUNK;    // 256
constexpr int VECS_PER_ROW     = COLS / 4;              // 32
}

// ---- CDNA5 async global -> LDS copy (ASYNCcnt tracked) ----------------------
__device__ __forceinline__ void async_copy_b128(const float* g, float* l) {
#if __has_builtin(__builtin_amdgcn_global_load_async_to_lds_b128)
  __builtin_amdgcn_global_load_async_to_lds_b128((AS1 v4i*)(void*)g,
                                                 (AS3 v4i*)(void*)l,
                                                 0, 0);
#else
  unsigned lds = (unsigned)(unsigned long long)(void*)l;     // low 32 bits = LDS byte offset
  asm volatile("global_load_async_to_lds_b128 %0, %1, off"
               :: "v"(lds), "v"((unsigned long long)g) : "memory");
#endif
}

template <int N>
__device__ __forceinline__ void wait_asynccnt() {
#if __has_builtin(__builtin_amdgcn_s_wait_asynccnt)
  __builtin_amdgcn_s_wait_asynccnt(N);
#else
  asm volatile("s_wait_asynccnt %0" :: "n"(N) : "memory");
#endif
}

// Stage one chunk of dW (rows [k*2*CHUNK, ...) of the [2*NSTEPS, BATCH] array,
// columns [c0, c0+COLS)) into LDS buffer bufIdx. All waves issue 8 b128 async
// loads each; tail rows are clamped (loaded redundantly, never read).
__device__ __forceinline__ void stage_chunk(const float* __restrict__ dW, float* lbuf,
                                            int bufIdx, int k, int c0, int tid) {
  float* ldst = lbuf + bufIdx * FLOATS_PER_CHUNK;
  const int baseRow = k * ROWS_PER_CHUNK;
#pragma unroll
  for (int j = 0; j < VECS_PER_THREAD; ++j) {
    const int vec  = j * COLS + tid;            // 0..1023, coalesced across lanes
    const int row  = vec / VECS_PER_ROW;        // = j*4 + tid/32
    const int colv = vec % VECS_PER_ROW;        // = tid%32
    int gRow = baseRow + row;
    if (gRow > 2 * NSTEPS - 1) gRow = 2 * NSTEPS - 1;
    const float* g = dW + (size_t)gRow * BATCH + c0 + (colv << 2);
    async_copy_b128(g, ldst + ((size_t)vec << 2));
  }
}

__global__ __launch_bounds__(COLS)
void sde_em_rollout(const float* __restrict__ y0,
                    const float* __restrict__ gammas,
                    const float* __restrict__ alphas,
                    const float* __restrict__ dW,
                    float* __restrict__ out) {
  __shared__ float lbuf[NBUF * FLOATS_PER_CHUNK];   // 96 KB of the WGP's 320 KB

  const int tid = threadIdx.x;
  const int c0  = blockIdx.x * COLS;
  const int col = c0 + tid;

  // initial state + out[0] = y0
  float yv0 = y0[col];
  float yv1 = y0[BATCH + col];
  out[col]         = yv0;
  out[BATCH + col] = yv1;

  // prime the async pipeline with NBUF-1 chunks
  for (int k = 0; k < NBUF - 1; ++k)
    stage_chunk(dW, lbuf, k, k, c0, tid);

  const float dt = (float)(624.0 / 4095.0);

  for (int k = 0; k < NCHUNK; ++k) {
    const int pk = k + NBUF - 1;
    if (pk < NCHUNK) {
      // buffer (k-1)%NBUF was released by the trailing barrier of iter k-1
      stage_chunk(dW, lbuf, pk % NBUF, pk, c0, tid);
      // retire the oldest of the NBUF outstanding chunk-groups (chunk k)
      wait_asynccnt<(NBUF - 1) * VECS_PER_THREAD>();
    } else {
      wait_asynccnt<0>();                 // drain phase
    }
    __syncthreads();                      // chunk k visible to all waves

    const float* buf = lbuf + (size_t)(k % NBUF) * FLOATS_PER_CHUNK;
    const int n0  = k * CHUNK;
    const int lim = (NSTEPS - n0 < CHUNK) ? (NSTEPS - n0) : CHUNK;
#pragma unroll 4
    for (int s = 0; s < lim; ++s) {
      const int n = n0 + s;
      const float ga  = gammas[n];        // uniform -> scalar cache load
      const float al  = alphas[n];
      const float dw0 = buf[(2 * s)     * COLS + tid];
      const float dw1 = buf[(2 * s + 1) * COLS + tid];

      // drift:  f0 = -y1 - |Q0 + Q1*(y0-B0)|*(y0-B0) ; f1 = (y0 + a*y1 - g)/TAU
      const float t  = __fsub_rn(yv0, 0.625f);
      const float q  = __fadd_rn(-9.0f, __fmul_rn(12.0f, t));
      const float f0 = __fsub_rn(-yv1, __fmul_rn(fabsf(q), t));
      const float f1 = __fdiv_rn(__fsub_rn(__fadd_rn(yv0, __fmul_rn(al, yv1)), ga),
                                 0.902f);
      // diffusion is constant sigma = 0.2 (mask terms cancel)
      yv0 = __fadd_rn(__fadd_rn(yv0, __fmul_rn(f0, dt)), __fmul_rn(0.2f, dw0));
      yv1 = __fadd_rn(__fadd_rn(yv1, __fmul_rn(f1, dt)), __fmul_rn(0.2f, dw1));

      float* o = out + (size_t)(n + 1) * (2 * BATCH) + col;
      __builtin_nontemporal_store(yv0, o);          // streaming, never re-read
      __builtin_nontemporal_store(yv1, o + BATCH);
    }
    __syncthreads();                      // release this buffer for overwrite
  }
}

extern "C" void kernel_launch(void* const* d_in, const int* in_sizes, int n_in,
                              void* d_out, int out_size, void* d_ws, size_t ws_size,
                              hipStream_t stream) {
  (void)in_sizes; (void)n_in; (void)out_size; (void)d_ws; (void)ws_size;
  const float* y0     = (const float*)d_in[0];   // [2, 16384]
  const float* gammas = (const float*)d_in[1];   // [4096]
  const float* alphas = (const float*)d_in[2];   // [4096]
  const float* dW     = (const float*)d_in[3];   // [4095, 2, 16384]
  float*       out    = (float*)d_out;           // [4096, 2, 16384]
  sde_em_rollout<<<NWG, COLS, 0, stream>>>(y0, gammas, alphas, dW, out);
}